// NonDiagLayer_25451976196773
// MI455X (gfx1250) — compile-verified
//
#include <hip/hip_runtime.h>
#include <hip/hip_bf16.h>

// ---------------------------------------------------------------------------
// MI455X (gfx1250). fp32 WMMA 16x16x4 for all per-edge GEMMs.
// Fragment-packed LDS layouts (one ds_load_b64 per operand fragment,
// bank-conflict-free), k-outer multi-tile GEMM for A-fragment reuse.
// 4-kernel fission; intermediates (<=100MB) stay L2-resident (192MB L2).
// ---------------------------------------------------------------------------

typedef __attribute__((ext_vector_type(2))) float v2f;
typedef __attribute__((ext_vector_type(8))) float v8f;

__device__ __forceinline__ float siluf(float x) { return x / (1.f + __expf(-x)); }
__device__ __forceinline__ float sspf(float x) {
  float ax = fabsf(x);
  return fmaxf(x, 0.f) + log1pf(__expf(-ax)) - 0.69314718055994530942f;
}

// Packed A element (row m in 0..15, col k): Ap[(k>>1)*32 + m*2 + (k&1)]
#define AIDX(row, col) ((((col) >> 1) << 5) + ((row) << 1) + ((col) & 1))
// Packed B element (row k, col n): Bp[(k>>1)*LDBP*2 + n*2 + (k&1)], LDBP in pairs

// Multi-tile GEMM, k-outer: NT 16x16 D tiles, B columns n0 + t*NSTRIDE + 0..15.
// A fragment (one ds_load_b64) reused across all NT tiles per k-step.
// LDBP chosen == 16 mod 32 so half-waves hit disjoint 32-bank sets.
template <int K, int LDBP, int NT, int NSTRIDE>
__device__ __forceinline__ void gemm_multi(const float* Ap, const float* Bp,
                                           int n0, v8f* acc) {
  const int lane = threadIdx.x & 31;
  const int m = lane & 15;
  const int half = lane >> 4;
  const float* arow = Ap + half * 32 + m * 2;
  const float* brow = Bp + (half * LDBP + n0 + m) * 2;
#pragma unroll
  for (int k4 = 0; k4 < K / 4; ++k4) {
    v2f a = *(const v2f*)(arow + k4 * 64);
#pragma unroll
    for (int t = 0; t < NT; ++t) {
      v2f b = *(const v2f*)(brow + (k4 * 2 * LDBP + t * NSTRIDE) * 2);
      acc[t] = __builtin_amdgcn_wmma_f32_16x16x4_f32(false, a, false, b,
                                                     (short)0, acc[t], false, false);
    }
  }
}

// Stage a KD x N row-major weight into fragment-packed LDS (pair-row pitch LDBP)
template <int KD, int N, int LDBP>
__device__ __forceinline__ void stage_B(const float* __restrict__ B, float* Bp,
                                        int tid) {
  for (int i = tid; i < (KD / 2) * N; i += 256) {
    int rp = i / N, n = i - rp * N;
    int k = rp * 2;
    Bp[(rp * LDBP + n) * 2]     = B[(size_t)k * N + n];
    Bp[(rp * LDBP + n) * 2 + 1] = B[(size_t)(k + 1) * N + n];
  }
}

// ---------------------------------------------------------------------------
// K1: per-node math (N=5000, tiny fraction of total FLOPs). VALU only.
// ---------------------------------------------------------------------------
__global__ void node_kernel(const float* __restrict__ x,
                            const float* __restrict__ Wli0, const float* __restrict__ Wli1,
                            const float* __restrict__ bli0,
                            const float* __restrict__ W1, const float* __restrict__ b1,
                            const float* __restrict__ W2, const float* __restrict__ b2,
                            const float* __restrict__ Wlp0, const float* __restrict__ Wlp1,
                            const float* __restrict__ blp0,
                            float* __restrict__ s0s, float* __restrict__ v0,
                            float* __restrict__ h, int N) {
  __shared__ float xs[256];
  __shared__ float cc[128];
  __shared__ float dd[128];
  __shared__ float gg[128];
  const int nid = blockIdx.x;
  if (nid >= N) return;
  const int t = threadIdx.x;
  xs[t] = x[(size_t)nid * 256 + t];
  __syncthreads();
  if (t < 64) {
    float acc = 0.f;
    for (int u = 0; u < 64; ++u) acc += xs[u] * Wli0[u * 64 + t];
    s0s[(size_t)nid * 64 + t] = acc * 0.125f + bli0[t];
    cc[t] = xs[t];
    float a0 = xs[64 + t * 3 + 0], a1 = xs[64 + t * 3 + 1], a2 = xs[64 + t * 3 + 2];
    cc[64 + t] = sqrtf(a0 * a0 + a1 * a1 + a2 * a2);
  } else {
    int i = t - 64, m = i >> 6, w = i & 63;
    float acc = 0.f;
    for (int u = 0; u < 64; ++u) acc += xs[64 + u * 3 + m] * Wli1[u * 64 + w];
    v0[(size_t)nid * 192 + w * 3 + m] = acc * 0.125f;
  }
  __syncthreads();
  if (t < 128) {
    float acc = b1[t];
    for (int j = 0; j < 128; ++j) acc += cc[j] * W1[j * 128 + t];
    dd[t] = siluf(acc);
  }
  __syncthreads();
  if (t < 128) {
    float acc = b2[t];
    for (int j = 0; j < 128; ++j) acc += dd[j] * W2[j * 128 + t];
    gg[t] = acc;
  }
  __syncthreads();
  if (t < 64) {
    float acc = 0.f;
    for (int u = 0; u < 64; ++u) acc += gg[u] * Wlp0[u * 64 + t];
    h[(size_t)nid * 256 + t] = acc * 0.125f + blp0[t];
  } else {
    int i = t - 64, m = i >> 6, w = i & 63;
    float acc = 0.f;
    for (int u = 0; u < 64; ++u) acc += xs[64 + u * 3 + m] * gg[64 + u] * Wlp1[u * 64 + w];
    h[(size_t)nid * 256 + 64 + w * 3 + m] = acc * 0.125f;
  }
}

// ---------------------------------------------------------------------------
// K2: per 16-edge wave tile: build s0 (packed), t=silu(s0@A1+a1b),
// q=ssp(ea@Wfc1/sqrt32); then per u-chunk j: 4 ws-tiles + 4 we-tiles,
// w = (we/sqrt8)*(ws+a2b), tensor-product epilogue -> u2, pair_v.
// ---------------------------------------------------------------------------
__global__ void edge_w_pair_kernel(const int* __restrict__ eidx_g,
                                   const float* __restrict__ ea,
                                   const float* __restrict__ s0s,
                                   const float* __restrict__ v0,
                                   const float* __restrict__ h,
                                   const float* __restrict__ Wfc1, const float* __restrict__ Wfc2,
                                   const float* __restrict__ A1, const float* __restrict__ a1b,
                                   const float* __restrict__ A2, const float* __restrict__ a2b,
                                   float* __restrict__ u2, float* __restrict__ pair_v, int E) {
  extern __shared__ float sm[];
  float* Wfc1s = sm;                // 256
  float* Wfc2p = Wfc1s + 256;       // 4 * 272 * 2 = 2176
  float* A1p   = Wfc2p + 2176;      // 64 * 80 * 2 = 10240
  float* A2p   = A1p + 10240;       // 32 * 272 * 2 = 17408
  float* a1bs  = A2p + 17408;       // 64
  float* a2bs  = a1bs + 64;         // 256
  float* pw    = a2bs + 256;        // 8 waves * 3232
  const int tid = threadIdx.x;

  for (int i = tid; i < 256; i += 256) Wfc1s[i] = Wfc1[i];
  stage_B<8, 256, 272>(Wfc2, Wfc2p, tid);
  stage_B<128, 64, 80>(A1, A1p, tid);
  stage_B<64, 256, 272>(A2, A2p, tid);
  if (tid < 64) a1bs[tid] = a1b[tid];
  for (int i = tid; i < 256; i += 256) a2bs[i] = a2b[i];
  __syncthreads();

  const int wv = tid >> 5, lane = tid & 31, m16 = lane & 15, half = lane >> 4;
  float* s0p = pw + wv * 3232;      // packed 16x128 = 2048
  float* tp  = s0p + 2048;          // packed 16x64  = 1024
  float* qp  = tp + 1024;           // packed 16x8   = 128
  int*   eixI = (int*)(qp + 128);   // dst[16], src[16]
  const int tile = blockIdx.x * 128 + wv * 16;

  if (lane < 16) {
    int e = tile + lane; if (e >= E) e = E - 1;
    eixI[lane]      = eidx_g[e];      // dst
    eixI[16 + lane] = eidx_g[E + e];  // src
  }
  // s0 stage, packed: lane = (row=m16, parity=half), cols 2i+half
  {
    int row = m16;
    int d = eixI[row], s = eixI[16 + row];
    const float* s0d = s0s + (size_t)d * 64;
    const float* s0r = s0s + (size_t)s * 64;
    const float* v0d = v0 + (size_t)d * 192;
    const float* v0r = v0 + (size_t)s * 192;
    for (int i = 0; i < 64; ++i) {
      int col = i * 2 + half;
      float val;
      if (col < 64) {
        val = 0.5f * (s0d[col] + s0r[col]);
      } else {
        int u = col - 64;
        val = (v0d[u * 3] * v0r[u * 3] + v0d[u * 3 + 1] * v0r[u * 3 + 1] +
               v0d[u * 3 + 2] * v0r[u * 3 + 2]) * (1.f / 3.f);
      }
      s0p[i * 32 + row * 2 + half] = val;  // AIDX(row, 2i+half)
    }
  }
  // q = ssp(ea @ Wfc1 / sqrt(32)), packed 16x8
  for (int j = 0; j < 4; ++j) {
    int idx = j * 32 + lane, row = idx >> 3, c = idx & 7;
    int e = tile + row; if (e >= E) e = E - 1;
    float acc = 0.f;
    for (int k = 0; k < 32; ++k) acc += ea[(size_t)e * 32 + k] * Wfc1s[k * 8 + c];
    qp[AIDX(row, c)] = sspf(acc * 0.17677669529663687f);
  }
  // t = silu(s0 @ A1 + a1b), packed 16x64
  {
    v8f tac[4] = {{}, {}, {}, {}};
    gemm_multi<128, 80, 4, 16>(s0p, A1p, 0, tac);
#pragma unroll
    for (int t4 = 0; t4 < 4; ++t4) {
      int col = t4 * 16 + m16;
#pragma unroll
      for (int r = 0; r < 8; ++r) {
        int row = (lane < 16) ? r : r + 8;
        tp[AIDX(row, col)] = siluf(tac[t4][r] + a1bs[col]);
      }
    }
  }
  // per u-chunk j: tiles {j, j+4, j+8, j+12} of w, then tensor-product epilogue
  const float ISQ8 = 0.35355339059327373f;
  const float CSS = 0.70710678118654752f;
  const float CVV = 0.40824829046386302f;
  const float CSV = 0.70710678118654752f;
#pragma unroll
  for (int j = 0; j < 4; ++j) {
    v8f aws[4] = {{}, {}, {}, {}};
    gemm_multi<64, 272, 4, 64>(tp, A2p, j * 16, aws);
    v8f awe[4] = {{}, {}, {}, {}};
    gemm_multi<8, 272, 4, 64>(qp, Wfc2p, j * 16, awe);
    const int u = j * 16 + m16;
    const float bb0 = a2bs[u], bb1 = a2bs[64 + u];
    const float bb2 = a2bs[128 + u], bb3 = a2bs[192 + u];
#pragma unroll
    for (int r = 0; r < 8; ++r) {
      int row = (lane < 16) ? r : r + 8;
      int e = tile + row;
      bool ok = (e < E);
      int eC = ok ? e : (E - 1);
      int d = eixI[row], s = eixI[16 + row];
      const float* hs = h + (size_t)s * 256;
      const float* hd = h + (size_t)d * 256;
      float wss = (awe[0][r] * ISQ8) * (aws[0][r] + bb0);
      float wsv = (awe[1][r] * ISQ8) * (aws[1][r] + bb1);
      float wvs = (awe[2][r] * ISQ8) * (aws[2][r] + bb2);
      float wvv = (awe[3][r] * ISQ8) * (aws[3][r] + bb3);
      float xsv = hs[u], ysv = hd[u];
      float dotv = 0.f, ov0, ov1, ov2;
      {
        float xv = hs[64 + u * 3 + 0], yv = hd[64 + u * 3 + 0];
        ov0 = CSV * (wsv * xsv * yv + wvs * xv * ysv); dotv += xv * yv;
      }
      {
        float xv = hs[64 + u * 3 + 1], yv = hd[64 + u * 3 + 1];
        ov1 = CSV * (wsv * xsv * yv + wvs * xv * ysv); dotv += xv * yv;
      }
      {
        float xv = hs[64 + u * 3 + 2], yv = hd[64 + u * 3 + 2];
        ov2 = CSV * (wsv * xsv * yv + wvs * xv * ysv); dotv += xv * yv;
      }
      float os = CSS * wss * xsv * ysv + CVV * wvv * dotv;
      float nn = sqrtf(ov0 * ov0 + ov1 * ov1 + ov2 * ov2);
      if (ok) {
        u2[(size_t)eC * 128 + u] = os;
        u2[(size_t)eC * 128 + 64 + u] = nn;
        pair_v[(size_t)eC * 192 + u * 3 + 0] = ov0;
        pair_v[(size_t)eC * 192 + u * 3 + 1] = ov1;
        pair_v[(size_t)eC * 192 + u * 3 + 2] = ov2;
      }
    }
  }
}

// ---------------------------------------------------------------------------
// K3: t2 = silu(u2 @ ng2_W1 + b1)   (E x 128 @ 128 x 128)
// ---------------------------------------------------------------------------
__global__ void ng2_gemm1_kernel(const float* __restrict__ u2,
                                 const float* __restrict__ W1, const float* __restrict__ b1,
                                 float* __restrict__ t2, int E) {
  extern __shared__ float sm[];
  float* W1p = sm;            // 64 * 144 * 2 = 18432
  float* b1s = W1p + 18432;   // 128
  float* pw  = b1s + 128;     // 8 * 2048
  const int tid = threadIdx.x;
  stage_B<128, 128, 144>(W1, W1p, tid);
  if (tid < 128) b1s[tid] = b1[tid];
  __syncthreads();
  const int wv = tid >> 5, lane = tid & 31, m16 = lane & 15, half = lane >> 4;
  float* Ap = pw + wv * 2048;   // packed 16x128
  const int tile = blockIdx.x * 128 + wv * 16;
  {
    int row = m16;
    int e = tile + row; if (e >= E) e = E - 1;
    const float* src = u2 + (size_t)e * 128;
    for (int i = 0; i < 32; ++i) {
      int col = i * 4 + half * 2;
      v2f vv = *(const v2f*)(src + col);
      *(v2f*)(Ap + (col >> 1) * 32 + row * 2) = vv;
    }
  }
  v8f acc[8] = {{}, {}, {}, {}, {}, {}, {}, {}};
  gemm_multi<128, 144, 8, 16>(Ap, W1p, 0, acc);
#pragma unroll
  for (int t8 = 0; t8 < 8; ++t8) {
    int col = t8 * 16 + m16;
#pragma unroll
    for (int r = 0; r < 8; ++r) {
      int row = (lane < 16) ? r : r + 8;
      int e = tile + row;
      if (e < E) t2[(size_t)e * 128 + col] = siluf(acc[t8][r] + b1s[col]);
    }
  }
}

// ---------------------------------------------------------------------------
// K4: g = t2 @ ng2_W2 + b2; gate pair_v by g[:,64:]; final e3nn_linear
// (g[:, :64]@Wr0, gated pair_v@Wr1 per component) + residual -> out.
// ---------------------------------------------------------------------------
__global__ void final_kernel(const float* __restrict__ t2,
                             const float* __restrict__ pair_v,
                             const float* __restrict__ npa,
                             const float* __restrict__ W2, const float* __restrict__ b2,
                             const float* __restrict__ Wr0, const float* __restrict__ Wr1,
                             const float* __restrict__ br0,
                             float* __restrict__ out, int E) {
  extern __shared__ float sm[];
  float* W2p  = sm;             // 64 * 144 * 2 = 18432
  float* Wr0p = W2p + 18432;    // 32 * 80 * 2 = 5120
  float* Wr1p = Wr0p + 5120;    // 5120
  float* b2s  = Wr1p + 5120;    // 128
  float* br0s = b2s + 128;      // 64
  float* pw   = br0s + 64;      // 8 * 5120
  const int tid = threadIdx.x;
  stage_B<128, 128, 144>(W2, W2p, tid);
  stage_B<64, 64, 80>(Wr0, Wr0p, tid);
  stage_B<64, 64, 80>(Wr1, Wr1p, tid);
  if (tid < 128) b2s[tid] = b2[tid];
  if (tid < 64)  br0s[tid] = br0[tid];
  __syncthreads();
  const int wv = tid >> 5, lane = tid & 31, m16 = lane & 15, half = lane >> 4;
  float* gA = pw + wv * 5120;   // packed 16x128 (g)
  float* rB = gA + 2048;        // 3072: t2 pack (2048) then gv 3 x packed 16x64
  const int tile = blockIdx.x * 128 + wv * 16;
  // stage t2 tile (packed)
  {
    int row = m16;
    int e = tile + row; if (e >= E) e = E - 1;
    const float* src = t2 + (size_t)e * 128;
    for (int i = 0; i < 32; ++i) {
      int col = i * 4 + half * 2;
      v2f vv = *(const v2f*)(src + col);
      *(v2f*)(rB + (col >> 1) * 32 + row * 2) = vv;
    }
  }
  // g = t2 @ W2 + b2 -> gA (packed)
  {
    v8f acc[8] = {{}, {}, {}, {}, {}, {}, {}, {}};
    gemm_multi<128, 144, 8, 16>(rB, W2p, 0, acc);
#pragma unroll
    for (int t8 = 0; t8 < 8; ++t8) {
      int col = t8 * 16 + m16;
#pragma unroll
      for (int r = 0; r < 8; ++r) {
        int row = (lane < 16) ? r : r + 8;
        gA[AIDX(row, col)] = acc[t8][r] + b2s[col];
      }
    }
  }
  // gated vector planes gv[m] (packed 16x64): pair_v[e][u][m] * g[e][64+u]
  {
    int row = m16;
    int e = tile + row; if (e >= E) e = E - 1;
    const float* pv = pair_v + (size_t)e * 192;
#pragma unroll
    for (int m = 0; m < 3; ++m) {
      for (int i = 0; i < 32; ++i) {
        int u = i * 2 + half;
        float gate = gA[(32 + i) * 32 + row * 2 + half];  // AIDX(row, 64+u)
        rB[m * 1024 + i * 32 + row * 2 + half] = pv[u * 3 + m] * gate;
      }
    }
  }
  // out_s = g[:, :64] @ Wr0 * 0.125 + br0 + npa
  {
    v8f aS[4] = {{}, {}, {}, {}};
    gemm_multi<64, 80, 4, 16>(gA, Wr0p, 0, aS);
#pragma unroll
    for (int t4 = 0; t4 < 4; ++t4) {
      int col = t4 * 16 + m16;
#pragma unroll
      for (int r = 0; r < 8; ++r) {
        int row = (lane < 16) ? r : r + 8;
        int e = tile + row;
        if (e < E) {
          size_t base = (size_t)e * 256;
          out[base + col] = aS[t4][r] * 0.125f + br0s[col] + npa[base + col];
        }
      }
    }
  }
  // out_v[m] = gv[m] @ Wr1 * 0.125 + npa
#pragma unroll
  for (int m = 0; m < 3; ++m) {
    v8f aV[4] = {{}, {}, {}, {}};
    gemm_multi<64, 80, 4, 16>(rB + m * 1024, Wr1p, 0, aV);
#pragma unroll
    for (int t4 = 0; t4 < 4; ++t4) {
      int col = t4 * 16 + m16;
#pragma unroll
      for (int r = 0; r < 8; ++r) {
        int row = (lane < 16) ? r : r + 8;
        int e = tile + row;
        if (e < E) {
          size_t base = (size_t)e * 256 + 64 + (size_t)col * 3 + m;
          out[base] = aV[t4][r] * 0.125f + npa[base];
        }
      }
    }
  }
}

// ---------------------------------------------------------------------------
extern "C" void kernel_launch(void* const* d_in, const int* in_sizes, int n_in,
                              void* d_out, int out_size, void* d_ws, size_t ws_size,
                              hipStream_t stream) {
  const float* node_attr = (const float*)d_in[0];
  const int*   eidx      = (const int*)  d_in[1];
  const float* ea        = (const float*)d_in[2];
  const float* npa       = (const float*)d_in[3];
  const float* Wli0 = (const float*)d_in[4];
  const float* Wli1 = (const float*)d_in[5];
  const float* bli0 = (const float*)d_in[6];
  const float* ng1W1 = (const float*)d_in[7];
  const float* ng1b1 = (const float*)d_in[8];
  const float* ng1W2 = (const float*)d_in[9];
  const float* ng1b2 = (const float*)d_in[10];
  const float* Wlp0 = (const float*)d_in[11];
  const float* Wlp1 = (const float*)d_in[12];
  const float* blp0 = (const float*)d_in[13];
  const float* Wfc1 = (const float*)d_in[14];
  const float* Wfc2 = (const float*)d_in[15];
  const float* A1   = (const float*)d_in[16];
  const float* a1b  = (const float*)d_in[17];
  const float* A2   = (const float*)d_in[18];
  const float* a2b  = (const float*)d_in[19];
  const float* ng2W1 = (const float*)d_in[20];
  const float* ng2b1 = (const float*)d_in[21];
  const float* ng2W2 = (const float*)d_in[22];
  const float* ng2b2 = (const float*)d_in[23];
  const float* Wr0 = (const float*)d_in[24];
  const float* Wr1 = (const float*)d_in[25];
  const float* br0 = (const float*)d_in[26];

  const int N = in_sizes[0] / 256;
  const int E = in_sizes[1] / 2;

  float* ws = (float*)d_ws;
  float* s0s    = ws;
  float* v0     = s0s + (size_t)N * 64;
  float* h      = v0 + (size_t)N * 192;
  float* u2     = h + (size_t)N * 256;
  float* pair_v = u2 + (size_t)E * 128;
  float* t2     = pair_v + (size_t)E * 192;

  node_kernel<<<N, 256, 0, stream>>>(node_attr, Wli0, Wli1, bli0,
                                     ng1W1, ng1b1, ng1W2, ng1b2,
                                     Wlp0, Wlp1, blp0, s0s, v0, h, N);

  const int eb = (E + 127) / 128;
  const size_t sm2 = (size_t)(256 + 2176 + 10240 + 17408 + 64 + 256 + 8 * 3232) * 4; // ~225KB
  edge_w_pair_kernel<<<eb, 256, sm2, stream>>>(eidx, ea, s0s, v0, h,
                                               Wfc1, Wfc2, A1, a1b, A2, a2b,
                                               u2, pair_v, E);

  const size_t sm3 = (size_t)(18432 + 128 + 8 * 2048) * 4;  // ~140KB
  ng2_gemm1_kernel<<<eb, 256, sm3, stream>>>(u2, ng2W1, ng2b1, t2, E);

  const size_t sm4 = (size_t)(18432 + 5120 + 5120 + 128 + 64 + 8 * 5120) * 4;  // ~279KB
  final_kernel<<<eb, 256, sm4, stream>>>(t2, pair_v, npa, ng2W2, ng2b2,
                                         Wr0, Wr1, br0, (float*)d_out, E);
}